// ModuleSepconv_27779848471328
// MI455X (gfx1250) — compile-verified
//
#include <hip/hip_runtime.h>
#include <cstdint>
#include <cstddef>

// Problem constants (match the JAX reference).
#define BB 8
#define CC 3
#define FF 5
#define HO 256
#define WO 256
#define WI (WO * FF)          // 1280 input width
#define HI (HO * FF)          // 1280 input height
#define ROWFLOATS (FF * WI)   // 6400 floats = 5 contiguous input rows
#define ROWVEC4   (ROWFLOATS / 4)  // 1600 float4 chunks (25600 B)

// out[b,c,y,x] = sum_{i,j} in[b,c,5y+i,5x+j] * V[b,i,y,x] * H[b,j,y,x]
// One block per (b, y); thread x = output column. Input rows for a block are
// contiguous (full-width rows 5y..5y+4), staged to LDS with CDNA5 async
// global->LDS copies tracked by ASYNCcnt.
__global__ __launch_bounds__(WO) void sepconv_bwlimited_kernel(
    const float* __restrict__ in,
    const float* __restrict__ vert,
    const float* __restrict__ horz,
    float* __restrict__ out)
{
    __shared__ __align__(16) float rows[ROWFLOATS];  // 25.6 KB

    const int x = threadIdx.x;   // 0..255
    const int y = blockIdx.x;    // 0..255
    const int b = blockIdx.y;    // 0..7

    // Per-pixel separable weights (coalesced loads, planes strided HO*WO).
    float v[FF], h[FF];
#pragma unroll
    for (int i = 0; i < FF; ++i) {
        const size_t pidx = (((size_t)b * FF + i) * HO + y) * WO + x;
        v[i] = vert[pidx];
        h[i] = horz[pidx];
    }

    // LDS byte address = low 32 bits of the generic shared pointer.
    const unsigned lds_base = (unsigned)(uintptr_t)&rows[0];

    for (int c = 0; c < CC; ++c) {
        // ---- async stage: 5 contiguous input rows (25600 B) -> LDS ----
        const float* gsrc =
            in + (((size_t)(b * CC + c) * HI + (size_t)y * FF) * WI);
#pragma unroll
        for (int t = 0; t < 7; ++t) {
            const int idx = t * 256 + x;           // float4 chunk index
            if (idx < ROWVEC4) {
                const float* gaddr = gsrc + (size_t)idx * 4;
                const unsigned laddr = lds_base + (unsigned)idx * 16u;
                asm volatile(
                    "global_load_async_to_lds_b128 %0, %1, off"
                    :
                    : "v"(laddr), "v"(gaddr)
                    : "memory");
            }
        }
        // Wait for this wave's async copies, then sync all waves' portions.
        asm volatile("s_wait_asynccnt 0x0" ::: "memory");
        __syncthreads();

        // ---- compute: acc = sum_i v[i] * (sum_j h[j] * row_i[5x+j]) ----
        float acc = 0.0f;
#pragma unroll
        for (int i = 0; i < FF; ++i) {
            const float* r = &rows[i * WI + x * FF];
            float dot = r[0] * h[0];
            dot = fmaf(r[1], h[1], dot);
            dot = fmaf(r[2], h[2], dot);
            dot = fmaf(r[3], h[3], dot);
            dot = fmaf(r[4], h[4], dot);
            acc = fmaf(v[i], dot, acc);
        }

        out[(((size_t)b * CC + c) * HO + y) * WO + x] = acc;

        // All lanes must finish reading LDS before next channel overwrites it.
        __syncthreads();
    }
}

extern "C" void kernel_launch(void* const* d_in, const int* in_sizes, int n_in,
                              void* d_out, int out_size, void* d_ws, size_t ws_size,
                              hipStream_t stream) {
    (void)in_sizes; (void)n_in; (void)d_ws; (void)ws_size; (void)out_size;
    const float* in   = (const float*)d_in[0];  // [B,C,HO*F,WO*F]
    const float* vert = (const float*)d_in[1];  // [B,F,HO,WO]
    const float* horz = (const float*)d_in[2];  // [B,F,HO,WO]
    float* out = (float*)d_out;                 // [B,C,HO,WO]

    dim3 grid(HO, BB);
    dim3 block(WO);
    sepconv_bwlimited_kernel<<<grid, block, 0, stream>>>(in, vert, horz, out);
}